// TreeCapsClassifier_764504178785
// MI455X (gfx1250) — compile-verified
//
#include <hip/hip_runtime.h>
#include <hip/hip_bf16.h>
#include <math.h>

typedef float v2f __attribute__((ext_vector_type(2)));
typedef float v8f __attribute__((ext_vector_type(8)));

#define NTOT 32768   // B*NPG nodes
#define BG   32      // graphs
#define NPGC 1024    // nodes per graph
#define HD   128     // feature size
#define NLAY 4
#define ACAP 8       // = BCAPS = 8 (top-k identical for a and b)
#define DCCC 16
#define NCLS 50
#define SCN  800     // NCLS*DCCC
#define NBLK 32      // n-blocks for dynamic-routing reduction
#define VTS_T 1024   // k_vts block size: 32 wave32 waves (max workgroup)

// ---------------- embedding gather ----------------
__global__ void k_embed(const int* __restrict__ tids, const int* __restrict__ kids,
                        const float* __restrict__ temb, const float* __restrict__ kemb,
                        float* __restrict__ h0) {
  int idx = blockIdx.x * blockDim.x + threadIdx.x;
  int n = idx >> 7, hh = idx & 127;
  float v = (hh < 64) ? temb[(size_t)tids[n] * 64 + hh]
                      : kemb[(size_t)kids[n] * 64 + (hh - 64)];
  h0[idx] = v;
}

// ---------------- CSR row ranges from sorted dst ----------------
__global__ void k_fill_rows(int* rs, int* re) {
  int i = blockIdx.x * blockDim.x + threadIdx.x;
  rs[i] = 0; re[i] = 0;
}
__global__ void k_build_rows(const int* __restrict__ dst, int* rs, int* re, int E) {
  int e = blockIdx.x * blockDim.x + threadIdx.x;
  if (e >= E) return;
  int d = dst[e];
  if (e == 0 || dst[e - 1] != d) rs[d] = e;
  if (e == E - 1 || dst[e + 1] != d) re[d] = e + 1;
}

// ---------------- triple GEMM via V_WMMA_F32_16X16X4_F32 ----------------
// O{0,1,2} = X @ W{0,1,2};  X:[NTOT,128], W:[128,128] row-major.
// Each wave computes one 16x16 C-tile for all three weight matrices.
__global__ void k_gemm3(const float* __restrict__ X,
                        const float* __restrict__ W0, const float* __restrict__ W1,
                        const float* __restrict__ W2,
                        float* __restrict__ O0, float* __restrict__ O1,
                        float* __restrict__ O2) {
  const int wave = threadIdx.x >> 5;     // 8 waves -> 8 column tiles
  const int lane = threadIdx.x & 31;
  const int half = lane >> 4;            // K split across lane halves
  const int id   = lane & 15;
  const int row0 = blockIdx.x * 16;
  const int col0 = wave * 16;

  v8f c0 = {0,0,0,0,0,0,0,0}, c1 = c0, c2 = c0;
  const float* xrow = X + (size_t)(row0 + id) * HD + 2 * half;

  for (int kk = 0; kk < HD / 4; ++kk) {
    const int k0 = kk * 4;
    // hint the streaming A operand ahead (emits global_prefetch_b8)
    if ((kk & 7) == 0) __builtin_prefetch(xrow + k0 + 32, 0, 1);
    v2f a;
    { float2 t = *(const float2*)(xrow + k0); a.x = t.x; a.y = t.y; }
    const int bi = (k0 + 2 * half) * HD + col0 + id;
    v2f b0, b1, b2;
    b0.x = W0[bi]; b0.y = W0[bi + HD];
    b1.x = W1[bi]; b1.y = W1[bi + HD];
    b2.x = W2[bi]; b2.y = W2[bi + HD];
    c0 = __builtin_amdgcn_wmma_f32_16x16x4_f32(false, a, false, b0, (short)0, c0, false, false);
    c1 = __builtin_amdgcn_wmma_f32_16x16x4_f32(false, a, false, b1, (short)0, c1, false, false);
    c2 = __builtin_amdgcn_wmma_f32_16x16x4_f32(false, a, false, b2, (short)0, c2, false, false);
  }
  size_t base = (size_t)(row0 + 8 * half) * HD + col0 + id;
#pragma unroll
  for (int v = 0; v < 8; ++v) {
    O0[base + (size_t)v * HD] = c0[v];
    O1[base + (size_t)v * HD] = c1[v];
    O2[base + (size_t)v * HD] = c2[v];
  }
}

// ---------------- per-node combine: child sum + relu + leaf passthrough ----
__global__ void k_combine(const float* __restrict__ hin, float* __restrict__ fout,
                          const float* __restrict__ GL, const float* __restrict__ GR,
                          const float* __restrict__ GT,
                          const int* __restrict__ src, const float* __restrict__ lw,
                          const float* __restrict__ rw,
                          const int* __restrict__ rs, const int* __restrict__ re,
                          const float* __restrict__ bconv) {
  int n = blockIdx.x, t = threadIdx.x;
  int e0 = rs[n], e1 = re[n];
  float old = hin[(size_t)n * HD + t];
  float out;
  if (e1 > e0) {
    float acc = GT[(size_t)n * HD + t] + bconv[t];
    for (int e = e0; e < e1; ++e) {
      int s = src[e];
      acc += GL[(size_t)s * HD + t] * lw[e] + GR[(size_t)s * HD + t] * rw[e];
    }
    out = fmaxf(acc, 0.f);
  } else {
    out = old;
  }
  fout[(size_t)n * HD + t] = out;
}

// ---------------- per-node L2 over H*L ----------------
__global__ void k_l2(const float* __restrict__ f, float* __restrict__ l2v) {
  __shared__ float red[HD];
  int n = blockIdx.x, t = threadIdx.x;
  float v = 0.f;
#pragma unroll
  for (int m = 0; m < NLAY; ++m) {
    float x = f[(size_t)m * NTOT * HD + (size_t)n * HD + t];
    v += x * x;
  }
  red[t] = v; __syncthreads();
  for (int off = HD / 2; off > 0; off >>= 1) {
    if (t < off) red[t] += red[t + off];
    __syncthreads();
  }
  if (t == 0) l2v[n] = red[0];
}

// ---------------- top-8 per graph (value desc, index asc on ties) --------
__global__ void k_top8(const float* __restrict__ l2v, int* __restrict__ topi) {
  __shared__ float vals[NPGC];
  __shared__ float rv[256];
  __shared__ int   ri[256];
  int b = blockIdx.x, t = threadIdx.x;
  for (int i = t; i < NPGC; i += 256) vals[i] = l2v[(size_t)b * NPGC + i];
  __syncthreads();
  for (int r = 0; r < ACAP; ++r) {
    float bv = -3.402823e38f; int bi = 0;
    for (int i = t; i < NPGC; i += 256) {
      float v = vals[i];
      if (v > bv) { bv = v; bi = i; }
    }
    rv[t] = bv; ri[t] = bi; __syncthreads();
    for (int off = 128; off > 0; off >>= 1) {
      if (t < off) {
        if (rv[t + off] > rv[t] || (rv[t + off] == rv[t] && ri[t + off] < ri[t])) {
          rv[t] = rv[t + off]; ri[t] = ri[t + off];
        }
      }
      __syncthreads();
    }
    if (t == 0) { topi[b * ACAP + r] = ri[0]; vals[ri[0]] = -3.402823e38f; }
    __syncthreads();
  }
}

// ---------------- gather u_i = uu[top] -> [b][1024][4] ----------------
__global__ void k_buildU(const float* __restrict__ f, const int* __restrict__ topi,
                         float* __restrict__ U) {
  int g = blockIdx.x;               // b*8 + cap
  int b = g >> 3, t = threadIdx.x;
  int node = topi[g];
#pragma unroll
  for (int m = 0; m < NLAY; ++m)
    U[((size_t)g * HD + t) * NLAY + m] =
        f[(size_t)m * NTOT * HD + ((size_t)b * NPGC + node) * HD + t];
}

// ---------------- VTS routing (rank-4 alpha trick), 32 waves/block ------
__global__ void k_vts(const float* __restrict__ U, float* __restrict__ SC) {
  __shared__ float u[NPGC * 4];
  __shared__ float vs[NPGC * 4];
  __shared__ float vn[NPGC * 4];
  __shared__ float mx[NPGC];
  __shared__ float zz[NPGC];
  int b = blockIdx.x, t = threadIdx.x;
  for (int i = t; i < NPGC * 4; i += VTS_T) {
    float v = U[(size_t)b * NPGC * 4 + i];
    u[i] = v; vs[i] = v;                       // Vsum starts at v0 == u
  }
  __syncthreads();
  for (int it = 0; it < 3; ++it) {
    // pass 1: per-row (i) online softmax stats over j of dot4(u[i], Vsum[j])
    {
      int i = t;  // exactly one row per thread
      float u0 = u[i*4], u1 = u[i*4+1], u2 = u[i*4+2], u3 = u[i*4+3];
      float m = -3.402823e38f, ssum = 0.f;
      for (int j = 0; j < NPGC; ++j) {
        float s = u0*vs[j*4] + u1*vs[j*4+1] + u2*vs[j*4+2] + u3*vs[j*4+3];
        if (s > m) { ssum = ssum * expf(m - s) + 1.f; m = s; }
        else ssum += expf(s - m);
      }
      mx[i] = m; zz[i] = ssum;
    }
    __syncthreads();
    // pass 2: v_new[j] = sum_i beta[i][j] * u[i]
    {
      int j = t;  // exactly one column per thread
      float w0 = vs[j*4], w1 = vs[j*4+1], w2 = vs[j*4+2], w3 = vs[j*4+3];
      float a0 = 0, a1 = 0, a2 = 0, a3 = 0;
      for (int i = 0; i < NPGC; ++i) {
        float s = u[i*4]*w0 + u[i*4+1]*w1 + u[i*4+2]*w2 + u[i*4+3]*w3;
        float beta = expf(s - mx[i]) / zz[i];
        a0 += beta * u[i*4];   a1 += beta * u[i*4+1];
        a2 += beta * u[i*4+2]; a3 += beta * u[i*4+3];
      }
      vn[j*4] = a0; vn[j*4+1] = a1; vn[j*4+2] = a2; vn[j*4+3] = a3;
    }
    __syncthreads();
    if (it < 2) {
      for (int i = t; i < NPGC * 4; i += VTS_T) vs[i] += vn[i];
      __syncthreads();
    }
  }
  // squash rows of v3 -> SC
  {
    int i = t;
    float x0 = vn[i*4], x1 = vn[i*4+1], x2 = vn[i*4+2], x3 = vn[i*4+3];
    float sq = x0*x0 + x1*x1 + x2*x2 + x3*x3;
    float sc = (sq / (1.f + sq)) / (sqrtf(sq + 1e-10f) + 1e-8f);
    size_t o = ((size_t)b * NPGC + i) * 4;
    SC[o] = x0 * sc; SC[o+1] = x1 * sc; SC[o+2] = x2 * sc; SC[o+3] = x3 * sc;
  }
}

// ---------------- dynamic routing: z partial sums ----------------
// gamma from zsum (delta == v . zsum); uniform==1 means gamma = 1/50.
__global__ void k_z(int uniform, const float* __restrict__ SC,
                    const float* __restrict__ Wjm, const float* __restrict__ zsum,
                    float* __restrict__ zpart) {
  __shared__ float zs[SCN];
  __shared__ float vb[SCN];
  __shared__ float tt[52];
  __shared__ float scn[4];
  int b = blockIdx.x >> 5, nb = blockIdx.x & 31, t = threadIdx.x;
  if (!uniform)
    for (int sc = t; sc < SCN; sc += 256) zs[sc] = zsum[(size_t)b * SCN + sc];
  float zacc[4] = {0.f, 0.f, 0.f, 0.f};
  __syncthreads();
  for (int nn = 0; nn < 32; ++nn) {
    int n = nb * 32 + nn;
    __syncthreads();
    if (t < 4) scn[t] = SC[((size_t)b * NPGC + n) * 4 + t];
    __syncthreads();
    for (int sc = t; sc < SCN; sc += 256) {
      int s = sc >> 4, c = sc & 15;
      const float4 w = *(const float4*)(Wjm + (((size_t)n * DCCC + c) * NCLS + s) * 4);
      vb[sc] = w.x * scn[0] + w.y * scn[1] + w.z * scn[2] + w.w * scn[3];
    }
    __syncthreads();
    if (!uniform) {
      if (t < NCLS) {
        float a = 0.f;
        for (int c = 0; c < DCCC; ++c) a += vb[t * 16 + c] * zs[t * 16 + c];
        tt[t] = a;
      }
      __syncthreads();
      if (t == 0) {
        float m = tt[0];
        for (int s = 1; s < NCLS; ++s) m = fmaxf(m, tt[s]);
        float d = 0.f;
        for (int s = 0; s < NCLS; ++s) d += expf(tt[s] - m);
        tt[50] = m; tt[51] = d;
      }
      __syncthreads();
    }
#pragma unroll
    for (int k = 0; k < 4; ++k) {
      int sc = t + k * 256;
      if (sc < SCN) {
        float g = uniform ? (1.f / 50.f) : expf(tt[sc >> 4] - tt[50]) / tt[51];
        zacc[k] += g * vb[sc];
      }
    }
  }
  __syncthreads();
#pragma unroll
  for (int k = 0; k < 4; ++k) {
    int sc = t + k * 256;
    if (sc < SCN) zpart[((size_t)b * NBLK + nb) * SCN + sc] = zacc[k];
  }
}

// ---------------- reduce zpart, squash, update zsum, emit logits --------
__global__ void k_zred(int first, int last, const float* __restrict__ zpart,
                       float* __restrict__ zsum, float* __restrict__ out) {
  __shared__ float zr[SCN];
  __shared__ float sscale[NCLS];
  int b = blockIdx.x, t = threadIdx.x;
  for (int sc = t; sc < SCN; sc += 256) {
    float acc = 0.f;
    for (int nb = 0; nb < NBLK; ++nb)
      acc += zpart[((size_t)b * NBLK + nb) * SCN + sc];
    zr[sc] = acc;
  }
  __syncthreads();
  if (t < NCLS) {
    float sq = 0.f;
    for (int c = 0; c < DCCC; ++c) { float x = zr[t * 16 + c]; sq += x * x; }
    float sc = (sq / (1.f + sq)) / (sqrtf(sq + 1e-10f) + 1e-8f);
    sscale[t] = sc;
    if (last) {
      float lg = sqrtf(sq * sc * sc + 1e-10f);
      out[b * NCLS + t] = lg;
      out[BG * NCLS + b * NCLS + t] = lg;   // tuple (logit, logit)
    }
  }
  __syncthreads();
  for (int sc = t; sc < SCN; sc += 256) {
    float zval = zr[sc] * sscale[sc >> 4];
    float base = first ? 0.f : zsum[(size_t)b * SCN + sc];
    zsum[(size_t)b * SCN + sc] = base + zval;
  }
}

extern "C" void kernel_launch(void* const* d_in, const int* in_sizes, int n_in,
                              void* d_out, int out_size, void* d_ws, size_t ws_size,
                              hipStream_t stream) {
  (void)n_in; (void)out_size; (void)ws_size;
  const int*   type_ids  = (const int*)d_in[0];
  const int*   token_ids = (const int*)d_in[1];
  const int*   src       = (const int*)d_in[2];
  const int*   dst       = (const int*)d_in[3];
  const float* lw        = (const float*)d_in[4];
  const float* rw        = (const float*)d_in[5];
  // d_in[6] has_children: unused (derived from CSR row ranges)
  const float* type_emb  = (const float*)d_in[7];
  const float* token_emb = (const float*)d_in[8];
  const float* W_left    = (const float*)d_in[9];
  const float* W_right   = (const float*)d_in[10];
  const float* W_top     = (const float*)d_in[11];
  const float* b_conv    = (const float*)d_in[12];
  const float* Wjm       = (const float*)d_in[13];
  float* out = (float*)d_out;
  const int E = in_sizes[2];

  const size_t NH = (size_t)NTOT * HD;
  float* h0   = (float*)d_ws;
  float* f    = h0 + NH;                 // 4 contiguous layer-feature planes
  float* GL   = f + 4 * NH;
  float* GR   = GL + NH;
  float* GT   = GR + NH;
  int*   rs   = (int*)(GT + NH);
  int*   re   = rs + NTOT;
  float* l2v  = (float*)(re + NTOT);
  int*   topi = (int*)(l2v + NTOT);
  float* U    = (float*)(topi + BG * ACAP);
  float* SCb  = U + (size_t)BG * NPGC * NLAY;
  float* zpart = SCb + (size_t)BG * NPGC * NLAY;
  float* zsum  = zpart + (size_t)BG * NBLK * SCN;

  k_embed<<<(NTOT * HD) / 256, 256, 0, stream>>>(type_ids, token_ids, type_emb, token_emb, h0);
  k_fill_rows<<<NTOT / 256, 256, 0, stream>>>(rs, re);
  k_build_rows<<<(E + 255) / 256, 256, 0, stream>>>(dst, rs, re, E);

  for (int l = 0; l < NLAY; ++l) {
    const float* hin = (l == 0) ? h0 : (f + (size_t)(l - 1) * NH);
    k_gemm3<<<NTOT / 16, 256, 0, stream>>>(hin,
        W_left + (size_t)l * HD * HD, W_right + (size_t)l * HD * HD,
        W_top + (size_t)l * HD * HD, GL, GR, GT);
    k_combine<<<NTOT, HD, 0, stream>>>(hin, f + (size_t)l * NH, GL, GR, GT,
                                       src, lw, rw, rs, re, b_conv + (size_t)l * HD);
  }

  k_l2<<<NTOT, HD, 0, stream>>>(f, l2v);
  k_top8<<<BG, 256, 0, stream>>>(l2v, topi);
  k_buildU<<<BG * ACAP, HD, 0, stream>>>(f, topi, U);
  k_vts<<<BG, VTS_T, 0, stream>>>(U, SCb);

  for (int r = 0; r < 3; ++r) {
    k_z<<<BG * NBLK, 256, 0, stream>>>(r == 0 ? 1 : 0, SCb, Wjm, zsum, zpart);
    k_zred<<<BG, 256, 0, stream>>>(r == 0 ? 1 : 0, r == 2 ? 1 : 0, zpart, zsum, out);
  }
}